// MetScore_70523363000464
// MI455X (gfx1250) — compile-verified
//
#include <hip/hip_runtime.h>
#include <math.h>

// ---------------------------------------------------------------------------
// MetScore for MI455X (gfx1250, wave32).
// HBM-bound streaming reduction: 110.6 MB @ 23.3 TB/s => ~4.7us floor.
//  - float4 non-temporal global loads (read-once stream)
//  - all threshold comparisons in normalized domain (monotone transform) so
//    the exponential only feeds smooth sums -> hardware v_exp_f32
//  - deterministic cross-lane fold via v_wmma_f32_16x16x4_f32 (A=ones,
//    B={acc,0} => D[m][n] = acc[n]+acc[n+16], mirrored across lane halves)
//  - mirrored fold stored by ALL lanes to LDS (no divergence, no bpermute),
//    row-summed with b128 LDS loads and scaled by 0.5
//  - fixed-order fp64 reduction of block partials in a tiny second kernel
// No atomics anywhere: bit-deterministic across replays.
// ---------------------------------------------------------------------------

typedef __attribute__((ext_vector_type(4))) float f4x;
typedef __attribute__((ext_vector_type(2))) float v2f;
typedef __attribute__((ext_vector_type(8))) float v8f;

#define NACC   26        // 6 correlation moments + 4 quantities * 5 levels
#define CHUNKS 48        // blocks per timestep
#define TSTEPS 20
#define SLAB4  57600     // 480*480/4 float4 per (b,t) slab
#define NB     2

__constant__ float d_LW[5] = {0.1f, 0.1f, 0.2f, 0.25f, 0.35f};
__constant__ float d_TW[20] = {0.0075f, 0.02f, 0.03f, 0.04f, 0.05f, 0.06f, 0.07f,
                               0.08f, 0.09f, 0.1f, 0.09f, 0.08f, 0.07f, 0.06f,
                               0.05f, 0.04f, 0.03f, 0.02f, 0.0075f, 0.005f};

// 32->16 lane fold, result mirrored across both 16-lane halves:
// returned value in lane j is v[j%16] + v[j%16+16].
__device__ __forceinline__ float wmma_fold(float v) {
#if defined(__gfx1250__) && __has_builtin(__builtin_amdgcn_wmma_f32_16x16x4_f32)
    v2f a; a.x = 1.0f; a.y = 1.0f;
    v2f b; b.x = v;    b.y = 0.0f;
    v8f c = {};
    v8f d = __builtin_amdgcn_wmma_f32_16x16x4_f32(
        false, a, false, b, (short)0, c, false, false);
    return d[0];
#else
    return v + __shfl_xor(v, 16, 32);
#endif
}

__global__ __launch_bounds__(256) void met_partial(
    const float* __restrict__ pred, const float* __restrict__ targ,
    const float* __restrict__ mask, float* __restrict__ ws) {
    const int t   = blockIdx.y;
    const int tid = threadIdx.x;
    const float LOGF = 3.4339872044851463f;       // log(30+1)
    // log1p(thr[l]) / log(31): threshold comparisons done on normalized input
    const float nthr[5] = {0.0277550f, 0.2018491f, 0.3199232f,
                           0.5217723f, 0.6398465f};

    float acc[NACC];
#pragma unroll
    for (int i = 0; i < NACC; ++i) acc[i] = 0.0f;

    const f4x* p4 = reinterpret_cast<const f4x*>(pred);
    const f4x* q4 = reinterpret_cast<const f4x*>(targ);
    const f4x* m4 = reinterpret_cast<const f4x*>(mask);

    const int V4 = NB * SLAB4;                    // float4 elems per timestep
    for (int v = blockIdx.x * 256 + tid; v < V4; v += CHUNKS * 256) {
        const int b   = (v >= SLAB4) ? 1 : 0;
        const int idx = (b * TSTEPS + t) * SLAB4 + (v - b * SLAB4);
        const f4x pv = __builtin_nontemporal_load(p4 + idx);
        const f4x qv = __builtin_nontemporal_load(q4 + idx);
        const f4x mv = __builtin_nontemporal_load(m4 + idx);
#pragma unroll
        for (int e = 0; e < 4; ++e) {
            const float pn = pv[e];
            const float qn = qv[e];
            const bool  M  = mv[e] > 0.5f;
            // physical values only feed smooth sums -> fast v_exp_f32 path
            float P = __expf(pn * LOGF) - 1.0f; P = fmaxf(P, 0.0f);
            float Q = __expf(qn * LOGF) - 1.0f; Q = fmaxf(Q, 0.0f);
            bool pge[5], qge[5];
#pragma unroll
            for (int l = 0; l < 5; ++l) {
                pge[l] = pn >= nthr[l];
                qge[l] = qn >= nthr[l];
            }
            const bool  dz  = !pge[0] && !qge[0];     // p<0.1 && t<0.1
            const float mcf = (M && !dz) ? 1.0f : 0.0f;
            acc[0] += mcf;
            acc[1] += P * mcf;
            acc[2] += Q * mcf;
            acc[3] += P * P * mcf;
            acc[4] += Q * Q * mcf;
            acc[5] += P * Q * mcf;
            const float diff = M ? fabsf(P - Q) : 0.0f;
#pragma unroll
            for (int l = 0; l < 5; ++l) {
                const bool pin = (l < 4) ? (pge[l] && !pge[l + 1]) : pge[4];
                const bool tin = (l < 4) ? (qge[l] && !qge[l + 1]) : qge[4];
                if (M) {
                    acc[6 + l]  += (pin && tin) ? 1.0f : 0.0f;
                    acc[11 + l] += pin ? 1.0f : 0.0f;
                    acc[16 + l] += tin ? 1.0f : 0.0f;
                    acc[21 + l] += tin ? diff : 0.0f;
                }
            }
        }
    }

    // Block reduction. WMMA fold per quantity; every lane stores its mirrored
    // fold value (EXEC all-1s throughout, no bpermute chains). Row q of smem
    // then holds each wave-sum twice -> final row sum * 0.5.
    __shared__ __align__(16) float smem[NACC][256];
#pragma unroll
    for (int q = 0; q < NACC; ++q) {
        smem[q][tid] = wmma_fold(acc[q]);
    }
    __syncthreads();
    if (tid < NACC) {
        const f4x* row = reinterpret_cast<const f4x*>(&smem[tid][0]);
        float s = 0.0f;
#pragma unroll
        for (int i = 0; i < 64; ++i) {
            const f4x r = row[i];
            s += r[0] + r[1] + r[2] + r[3];
        }
        ws[(t * CHUNKS + blockIdx.x) * NACC + tid] = 0.5f * s;
    }
}

__global__ __launch_bounds__(256) void met_final(const float* __restrict__ ws,
                                                 float* __restrict__ out) {
    __shared__ double red[TSTEPS][NACC];
    __shared__ float ts_s[TSTEPS][5], mae_s[TSTEPS][5], r_s[TSTEPS], sc_s[TSTEPS];
    const int tid = threadIdx.x;

    // Fixed-order reduction of block partials (deterministic).
    for (int task = tid; task < TSTEPS * NACC; task += 256) {
        const int t = task / NACC, q = task % NACC;
        double s = 0.0;
        for (int c = 0; c < CHUNKS; ++c)
            s += (double)ws[(t * CHUNKS + c) * NACC + q];
        red[t][q] = s;
    }
    __syncthreads();

    if (tid < TSTEPS) {
        const int t = tid;
        const double cnt = red[t][0], sp = red[t][1], st = red[t][2];
        const double spp = red[t][3], stt = red[t][4], spt = red[t][5];
        // Centered moments from raw moments (one-pass formula, exact algebra).
        double r = 0.0;
        if (cnt > 0.0) {
            const double num = spt - sp * st / cnt;
            const double vp = fmax(spp - sp * sp / cnt, 0.0);
            const double vt = fmax(stt - st * st / cnt, 0.0);
            r = num / (sqrt(vp * vt) + 1e-6);
            r = fmin(fmax(r, -1.0), 1.0);
        }
        r_s[t] = (float)r;
        const float term_corr = sqrtf(expf((float)r - 1.0f));
        float sc = 0.0f;
#pragma unroll
        for (int l = 0; l < 5; ++l) {
            const double hits = red[t][6 + l];
            const double ptot = red[t][11 + l];
            const double ttot = red[t][16 + l];
            const double mnum = red[t][21 + l];
            const float ts  = (float)(hits / (ptot + ttot - hits + 1e-8));
            const float mae = (ttot > 0.0) ? (float)(mnum / fmax(ttot, 1.0)) : 0.0f;
            ts_s[t][l]  = ts;
            mae_s[t][l] = mae;
            sc += d_LW[l] * ts * sqrtf(expf(-mae * 0.01f));
        }
        sc_s[t] = term_corr * sc;
    }
    __syncthreads();

    // Outputs: [0]=total, [1..20]=score_time, [21..40]=r_time,
    // [41..140]=ts_mat, [141..240]=mae_mat, [241..245]=ts_mean, [246..250]=mae_mean
    if (tid == 0) {
        float tot = 0.0f;
        for (int t = 0; t < TSTEPS; ++t) tot += sc_s[t] * d_TW[t];
        out[0] = tot;
    }
    for (int i = tid; i < TSTEPS; i += 256) {
        out[1 + i]  = sc_s[i];
        out[21 + i] = r_s[i];
    }
    for (int i = tid; i < TSTEPS * 5; i += 256) {
        out[41 + i]  = ts_s[i / 5][i % 5];
        out[141 + i] = mae_s[i / 5][i % 5];
    }
    if (tid < 5) {
        double a = 0.0, b = 0.0;
        for (int t = 0; t < TSTEPS; ++t) { a += ts_s[t][tid]; b += mae_s[t][tid]; }
        out[241 + tid] = (float)(a / TSTEPS);
        out[246 + tid] = (float)(b / TSTEPS);
    }
}

extern "C" void kernel_launch(void* const* d_in, const int* in_sizes, int n_in,
                              void* d_out, int out_size, void* d_ws, size_t ws_size,
                              hipStream_t stream) {
    const float* pred = (const float*)d_in[0];
    const float* targ = (const float*)d_in[1];
    const float* mask = (const float*)d_in[2];
    float* out = (float*)d_out;
    float* ws  = (float*)d_ws;   // needs 20*48*26*4 = 99,840 bytes

    dim3 grid(CHUNKS, TSTEPS);
    met_partial<<<grid, 256, 0, stream>>>(pred, targ, mask, ws);
    met_final<<<1, 256, 0, stream>>>(ws, out);
}